// DCTLayer_3307124818228
// MI455X (gfx1250) — compile-verified
//
#include <hip/hip_runtime.h>
#include <math.h>

// DCT-II 2D layer: Y = D * X * D^T, then log(|Y|+eps), then (y-mean)/std.
// B=128, C=3, H=W=256.  One workgroup per (b,c) image, 8 waves (wave32).
// fp32 WMMA (V_WMMA_F32_16X16X4_F32) keeps full reference precision; the
// whole input is L2-resident (100MB < 192MB L2) so global A/B re-reads are cheap.

#define N       256
#define NIMG    (128 * 3)
#define LDT     260           // padded LDS row stride (floats): 260 % 64 != 0 -> no bank aliasing
#define EPSV    1e-13f
#define PI_D    3.14159265358979323846

typedef __attribute__((ext_vector_type(2))) float v2f;
typedef __attribute__((ext_vector_type(8))) float v8f;

// ---------------------------------------------------------------------------
// Kernel 0: materialize orthonormal DCT-II basis D[k,m] into d_ws (256KB).
// Computed in f64 (65536 elements, negligible cost) for exact basis values.
// ---------------------------------------------------------------------------
__global__ void build_dct_kernel(float* __restrict__ D) {
    int idx = blockIdx.x * blockDim.x + threadIdx.x;
    if (idx >= N * N) return;
    int k = idx >> 8;
    int m = idx & 255;
    double s   = (k == 0) ? sqrt(1.0 / (double)N) : sqrt(2.0 / (double)N);
    double ang = PI_D * (double)((2 * m + 1) * k) / (2.0 * (double)N);
    D[idx] = (float)(s * cos(ang));
}

// ---------------------------------------------------------------------------
// Main kernel: per-image fused  T = X*D^T (LDS)  ->  Y = D*T  -> log/normalize
// ---------------------------------------------------------------------------
__global__ __launch_bounds__(256) void dct2d_wmma_kernel(
        const float* __restrict__ X,      // [B,C,256,256]
        const float* __restrict__ meanv,  // [C]
        const float* __restrict__ stdv,   // [C]
        const float* __restrict__ D,      // [256,256] DCT basis
        float* __restrict__ Out)          // [B,C,256,256]
{
    __shared__ float sT[N * LDT];         // 266,240 B of the 320KB WGP LDS

    const int img = blockIdx.x;
    const int c   = img % 3;
    const float* x   = X   + (size_t)img * (N * N);
    float*       out = Out + (size_t)img * (N * N);

    const int tid  = threadIdx.x;
    const int wave = tid >> 5;            // 0..7
    const int lane = tid & 31;
    const int half = lane >> 4;           // 0/1: selects K-pair (A/B) or M+8 (C/D)
    const int l16  = lane & 15;

    // =========================  Phase 1:  T = X * D^T  =======================
    // T[k,l] = sum_w X[k,w] * D[l,w].  Wave owns row-tiles kt = 2*wave..2*wave+1,
    // keeps all 16 column-tile accumulators live (16 x v8f = 128 VGPRs).
    #pragma unroll 1
    for (int ktc = 0; ktc < 2; ++ktc) {
        const int kt = wave * 2 + ktc;
        v8f acc[16];
        #pragma unroll
        for (int i = 0; i < 16; ++i) acc[i] = (v8f)(0.0f);

        #pragma unroll 1
        for (int wc = 0; wc < 16; ++wc) {           // w-chunk of 16 (4 WMMA k-steps)
            const int w0 = wc * 16;
            v2f afrag[4];                           // X fragments, reused for all 16 lt
            #pragma unroll
            for (int s = 0; s < 4; ++s) {
                const int kk = w0 + s * 4 + 2 * half;
                const float2 t = *reinterpret_cast<const float2*>(
                    x + (size_t)(kt * 16 + l16) * N + kk);
                afrag[s].x = t.x; afrag[s].y = t.y;
            }
            #pragma unroll
            for (int lt = 0; lt < 16; ++lt) {
                #pragma unroll
                for (int s = 0; s < 4; ++s) {
                    const int kk = w0 + s * 4 + 2 * half;
                    // B[w,n] = D[lt*16+n, w] -> contiguous float2 in w
                    const float2 t = *reinterpret_cast<const float2*>(
                        D + (size_t)(lt * 16 + l16) * N + kk);
                    v2f b; b.x = t.x; b.y = t.y;
                    acc[lt] = __builtin_amdgcn_wmma_f32_16x16x4_f32(
                        false, afrag[s], false, b, (short)0, acc[lt], false, false);
                }
            }
        }
        // Spill T tiles to padded LDS (lanes 0-15 write 16 consecutive floats).
        #pragma unroll
        for (int lt = 0; lt < 16; ++lt) {
            #pragma unroll
            for (int r = 0; r < 8; ++r) {
                sT[(kt * 16 + r + 8 * half) * LDT + lt * 16 + l16] = acc[lt][r];
            }
        }
    }

    __syncthreads();

    const float mu     = meanv[c];
    const float inv_sd = 1.0f / stdv[c];

    // =========================  Phase 2:  Y = D * T  =========================
    // Y[h,l] = sum_k D[h,k] * T[k,l]; A from D (global, L2/L0-hot), B from LDS.
    #pragma unroll 1
    for (int htc = 0; htc < 2; ++htc) {
        const int ht = wave * 2 + htc;
        v8f acc[16];
        #pragma unroll
        for (int i = 0; i < 16; ++i) acc[i] = (v8f)(0.0f);

        #pragma unroll 1
        for (int kc = 0; kc < 16; ++kc) {
            const int k0 = kc * 16;
            v2f afrag[4];
            #pragma unroll
            for (int s = 0; s < 4; ++s) {
                const int kk = k0 + s * 4 + 2 * half;
                const float2 t = *reinterpret_cast<const float2*>(
                    D + (size_t)(ht * 16 + l16) * N + kk);
                afrag[s].x = t.x; afrag[s].y = t.y;
            }
            #pragma unroll
            for (int lt = 0; lt < 16; ++lt) {
                #pragma unroll
                for (int s = 0; s < 4; ++s) {
                    const int kk = k0 + s * 4 + 2 * half;
                    v2f b;
                    b.x = sT[(kk    ) * LDT + lt * 16 + l16];
                    b.y = sT[(kk + 1) * LDT + lt * 16 + l16];
                    acc[lt] = __builtin_amdgcn_wmma_f32_16x16x4_f32(
                        false, afrag[s], false, b, (short)0, acc[lt], false, false);
                }
            }
        }
        // Fused epilogue: log(|y|+eps), normalize, coalesced 16-wide stores.
        #pragma unroll
        for (int lt = 0; lt < 16; ++lt) {
            #pragma unroll
            for (int r = 0; r < 8; ++r) {
                const float v = acc[lt][r];
                const float y = logf(fabsf(v) + EPSV);
                out[(size_t)(ht * 16 + r + 8 * half) * N + lt * 16 + l16] =
                    (y - mu) * inv_sd;
            }
        }
    }
}

// ---------------------------------------------------------------------------
extern "C" void kernel_launch(void* const* d_in, const int* in_sizes, int n_in,
                              void* d_out, int out_size, void* d_ws, size_t ws_size,
                              hipStream_t stream) {
    const float* x    = (const float*)d_in[0];   // [128,3,256,256] f32
    const float* mean = (const float*)d_in[1];   // [3] f32
    const float* stdv = (const float*)d_in[2];   // [3] f32
    float*       out  = (float*)d_out;
    float*       D    = (float*)d_ws;            // needs 256*256*4 = 256KB scratch

    (void)in_sizes; (void)n_in; (void)out_size; (void)ws_size;

    build_dct_kernel<<<(N * N + 255) / 256, 256, 0, stream>>>(D);
    dct2d_wmma_kernel<<<NIMG, 256, 0, stream>>>(x, mean, stdv, D, out);
}